// NER2IC_8564164788270
// MI455X (gfx1250) — compile-verified
//
#include <hip/hip_runtime.h>

typedef __attribute__((ext_vector_type(2))) float v2f;
typedef __attribute__((ext_vector_type(8))) float v8f;

#define B_ 32
#define S_ 128
#define D_ 1024
#define H_ 512
#define C_ 64
#define EPS_ 1e-5f

// ---------------------------------------------------------------------------
// K1: ner_mean[b,d] = mean_s ner[b,s,d]   (coalesced along d)
// ---------------------------------------------------------------------------
__global__ void k_ner_mean(const float* __restrict__ ner, float* __restrict__ nm) {
    int gid = blockIdx.x * blockDim.x + threadIdx.x;   // 0 .. B*D-1
    int b = gid >> 10;            // / D
    int d = gid & (D_ - 1);
    const float* p = ner + (size_t)b * S_ * D_ + d;
    float s = 0.f;
#pragma unroll 8
    for (int ss = 0; ss < S_; ++ss) s += p[(size_t)ss * D_];
    nm[gid] = s * (1.0f / S_);
}

// ---------------------------------------------------------------------------
// K2: v[32,512] = nm[32,1024] @ W1[1024,512]   via V_WMMA_F32_16X16X4_F32
// One wave per 16x16 output tile. grid = 2*32 = 64 waves.
// A frag (16x4 f32, 2 VGPR): lanes 0-15 -> K=0,1 ; lanes 16-31 -> K=2,3
// B frag (4x16 f32, 2 VGPR): N striped over lanes, same K split.
// C/D (8 VGPR): lanes 0-15 M=j, lanes 16-31 M=8+j, N=lane&15.
// ---------------------------------------------------------------------------
__global__ void k_gemm_v(const float* __restrict__ A, const float* __restrict__ Bm,
                         float* __restrict__ Cm) {
    const int lane = threadIdx.x;
    const int half = lane >> 4;
    const int r    = lane & 15;
    const int mt = blockIdx.x >> 5;   // 0..1
    const int nt = blockIdx.x & 31;   // 0..31
    const int m0 = mt * 16, n0 = nt * 16;
    v8f acc = {};
    for (int k0 = 0; k0 < D_; k0 += 4) {
        const int ka = k0 + 2 * half;
        v2f a, bf;
        a.x  = A[(m0 + r) * D_ + ka];
        a.y  = A[(m0 + r) * D_ + ka + 1];
        bf.x = Bm[ka * H_ + n0 + r];
        bf.y = Bm[(ka + 1) * H_ + n0 + r];
        acc = __builtin_amdgcn_wmma_f32_16x16x4_f32(
            /*neg_a=*/false, a, /*neg_b=*/false, bf,
            /*c_mod=*/(short)0, acc, /*reuse_a=*/false, /*reuse_b=*/false);
    }
#pragma unroll
    for (int j = 0; j < 8; ++j) {
        const int row = m0 + half * 8 + j;
        const int col = n0 + r;
        Cm[row * H_ + col] = acc[j];
    }
}

// ---------------------------------------------------------------------------
// K3: per-b LayerNorm moments: mv=E[v], mb=E[b1], Svv=Var(v), Svb=Cov(v,b1),
//     Sbb=Var(b1).  One block per b.
// ---------------------------------------------------------------------------
__global__ void k_stats(const float* __restrict__ v, const float* __restrict__ b1,
                        float* __restrict__ stats) {
    __shared__ float red[5][256];
    const int b = blockIdx.x, tid = threadIdx.x;
    float sv = 0, svv = 0, svb = 0, sb = 0, sbb = 0;
    for (int h = tid; h < H_; h += 256) {
        const float vv = v[b * H_ + h];
        const float bb = b1[h];
        sv += vv; svv += vv * vv; svb += vv * bb; sb += bb; sbb += bb * bb;
    }
    red[0][tid] = sv; red[1][tid] = svv; red[2][tid] = svb;
    red[3][tid] = sb; red[4][tid] = sbb;
    __syncthreads();
    for (int off = 128; off > 0; off >>= 1) {
        if (tid < off) {
#pragma unroll
            for (int j = 0; j < 5; ++j) red[j][tid] += red[j][tid + off];
        }
        __syncthreads();
    }
    if (tid == 0) {
        const float inv = 1.0f / H_;
        const float mv = red[0][0] * inv, mb = red[3][0] * inv;
        float* st = stats + b * 8;
        st[0] = mv;
        st[1] = mb;
        st[2] = red[1][0] * inv - mv * mv;   // Svv
        st[3] = red[2][0] * inv - mv * mb;   // Svb
        st[4] = red[4][0] * inv - mb * mb;   // Sbb
    }
}

// ---------------------------------------------------------------------------
// K4: pooled[b,h] = (1/D) * sum_i gelu( (a_i*(v_h-mv) + (b1_h-mb)) * r_i * g_h + beta_h )
//     with r_i = rsqrt(a_i^2*Svv + 2 a_i*Svb + Sbb + eps), a_i = ic[b,i].
//     Precompute p_i = a_i*r_i, q_i = r_i into LDS, then a pure FMA+erf loop:
//     y = vg*p_i + bg*q_i + beta_h.
// ---------------------------------------------------------------------------
__global__ void k_pool(const float* __restrict__ ic, const float* __restrict__ v,
                       const float* __restrict__ b1, const float* __restrict__ gamma,
                       const float* __restrict__ beta, const float* __restrict__ stats,
                       float* __restrict__ pooled) {
    __shared__ float p[D_];
    __shared__ float q[D_];
    const int b = blockIdx.x, tid = threadIdx.x;
    const float* st = stats + b * 8;
    const float mv = st[0], mb = st[1], Svv = st[2], Svb = st[3], Sbb = st[4];
    for (int i = tid; i < D_; i += 256) {
        const float a   = ic[b * D_ + i];
        const float var = fmaf(a, fmaf(a, Svv, 2.0f * Svb), Sbb) + EPS_;
        const float rr  = rsqrtf(var);
        p[i] = a * rr;
        q[i] = rr;
    }
    __syncthreads();
    for (int h = tid; h < H_; h += 256) {
        const float g  = gamma[h];
        const float vg = (v[b * H_ + h] - mv) * g;
        const float bg = (b1[h] - mb) * g;
        const float be = beta[h];
        float acc = 0.f;
        for (int i = 0; i < D_; ++i) {
            const float y = fmaf(vg, p[i], fmaf(bg, q[i], be));
            acc += 0.5f * y * (1.0f + erff(y * 0.70710678118654752f)); // exact GELU
        }
        pooled[b * H_ + h] = acc * (1.0f / D_);
    }
}

// ---------------------------------------------------------------------------
// K5: out[32,64] = pooled[32,512] @ W2[512,64] + b2   via fp32 WMMA.
// grid = 2*4 = 8 waves.
// ---------------------------------------------------------------------------
__global__ void k_gemm_out(const float* __restrict__ A, const float* __restrict__ Bm,
                           const float* __restrict__ b2, float* __restrict__ Out) {
    const int lane = threadIdx.x;
    const int half = lane >> 4;
    const int r    = lane & 15;
    const int mt = blockIdx.x >> 2;   // 0..1
    const int nt = blockIdx.x & 3;    // 0..3
    const int m0 = mt * 16, n0 = nt * 16;
    v8f acc = {};
    for (int k0 = 0; k0 < H_; k0 += 4) {
        const int ka = k0 + 2 * half;
        v2f a, bf;
        a.x  = A[(m0 + r) * H_ + ka];
        a.y  = A[(m0 + r) * H_ + ka + 1];
        bf.x = Bm[ka * C_ + n0 + r];
        bf.y = Bm[(ka + 1) * C_ + n0 + r];
        acc = __builtin_amdgcn_wmma_f32_16x16x4_f32(
            false, a, false, bf, (short)0, acc, false, false);
    }
#pragma unroll
    for (int j = 0; j < 8; ++j) {
        const int row = m0 + half * 8 + j;
        const int col = n0 + r;
        Out[row * C_ + col] = acc[j] + b2[col];
    }
}

// ---------------------------------------------------------------------------
extern "C" void kernel_launch(void* const* d_in, const int* in_sizes, int n_in,
                              void* d_out, int out_size, void* d_ws, size_t ws_size,
                              hipStream_t stream) {
    const float* ic    = (const float*)d_in[0];  // [32,1024]
    const float* ner   = (const float*)d_in[1];  // [32,128,1024]
    const float* W1    = (const float*)d_in[2];  // [1024,512]
    const float* b1    = (const float*)d_in[3];  // [512]
    const float* gamma = (const float*)d_in[4];  // [512]
    const float* beta  = (const float*)d_in[5];  // [512]
    const float* W2    = (const float*)d_in[6];  // [512,64]
    const float* b2    = (const float*)d_in[7];  // [64]
    float* out = (float*)d_out;                  // [32,64]

    float* ws     = (float*)d_ws;
    float* nm     = ws;              // 32*1024 = 32768 floats
    float* v      = ws + 32768;      // 32*512  = 16384
    float* stats  = ws + 49152;      // 32*8    = 256
    float* pooled = ws + 49408;      // 32*512  = 16384   (total ~258 KB)

    k_ner_mean<<<(B_ * D_) / 256, 256, 0, stream>>>(ner, nm);
    k_gemm_v  <<<64, 32, 0, stream>>>(nm, W1, v);
    k_stats   <<<B_, 256, 0, stream>>>(v, b1, stats);
    k_pool    <<<B_, 256, 0, stream>>>(ic, v, b1, gamma, beta, stats, pooled);
    k_gemm_out<<<8, 32, 0, stream>>>(pooled, W2, b2, out);
}